// ContextualRoIAlign_8469675507724
// MI455X (gfx1250) — compile-verified
//
#include <hip/hip_runtime.h>
#include <hip/hip_bf16.h>

// ---- problem constants (fixed by setup_inputs) ----
#define BATCH   2
#define CTOT    256
#define DIMHW   56
#define NBOX    64
#define MGT     8
#define PP      7
#define RPI     (NBOX + NBOX*MGT)      // 576 rois per image
#define AMAT_FL (2*16*56)              // Ay[16][56] + Ax[16][56] floats per roi
#define FMT_FL  ((size_t)BATCH*CTOT*64*56)  // padded transposed fm, floats
#define CB      8                      // channels per workgroup
#define CBLK    (CTOT/CB)              // 32

typedef __attribute__((ext_vector_type(2))) float v2f;
typedef __attribute__((ext_vector_type(8))) float v8f;

__device__ __forceinline__ v8f wmma4(v2f a, v2f b, v8f c) {
  // D = A(16x4) * B(4x16) + C, f32 everywhere.
  return __builtin_amdgcn_wmma_f32_16x16x4_f32(
      /*neg_a=*/false, a, /*neg_b=*/false, b,
      /*c_mod=*/(short)0, c, /*reuse_a=*/false, /*reuse_b=*/false);
}

// ---------------------------------------------------------------------------
// Kernel 1: fmT[b][c][w(0..63)][h(0..55)] = (w<56) ? fm[b][c][h][w] : 0
// ---------------------------------------------------------------------------
__global__ __launch_bounds__(256) void transpose_fm_kernel(
    const float* __restrict__ fm, float* __restrict__ fmT) {
  const int bc = blockIdx.x;                       // 0 .. B*C-1
  const float* src = fm + (size_t)bc * (DIMHW*DIMHW);
  float* dst = fmT + (size_t)bc * (64*DIMHW);
  for (int i = threadIdx.x; i < 64*DIMHW; i += blockDim.x) {
    const int w = i / DIMHW;
    const int h = i % DIMHW;
    dst[i] = (w < DIMHW) ? src[h*DIMHW + w] : 0.0f;
  }
}

// ---------------------------------------------------------------------------
// Kernel 2: per-roi interpolation matrices.
//   roi r: b = r/RPI, j = r%RPI. j<64 -> box roi (scale 1); else union box
//   (n = (j-64)/8, m = (j-64)%8) with scale 1/M baked into Ax.
//   Layout per roi: Ay[16][56] then Ax[16][56]; rows 7..15 zero.
// ---------------------------------------------------------------------------
__global__ __launch_bounds__(64) void prep_axes_kernel(
    const float* __restrict__ boxes,   // [B][64][4]
    const float* __restrict__ gts,     // [B][8][4]
    float* __restrict__ Amats) {
  const int r = blockIdx.x;
  const int b = r / RPI;
  const int j = r % RPI;

  float x1, y1, x2, y2, scale;
  if (j < NBOX) {
    const float* bb = boxes + ((size_t)(b*NBOX + j)) * 4;
    x1 = bb[0]; y1 = bb[1]; x2 = bb[2]; y2 = bb[3];
    scale = 1.0f;
  } else {
    const int jj = j - NBOX;
    const int n = jj / MGT, m = jj % MGT;
    const float* bb = boxes + ((size_t)(b*NBOX + n)) * 4;
    const float* gg = gts   + ((size_t)(b*MGT  + m)) * 4;
    x1 = fminf(bb[0], gg[0]); y1 = fminf(bb[1], gg[1]);
    x2 = fmaxf(bb[2], gg[2]); y2 = fmaxf(bb[3], gg[3]);
    scale = 1.0f / (float)MGT;
  }

  float* base = Amats + (size_t)r * AMAT_FL;
  const int t = threadIdx.x;

  if (t < 2*PP) {
    const int axis = t / PP;            // 0 = Ay (y), 1 = Ax (x)
    const int p    = t % PP;
    const float start = axis ? x1 : y1;
    const float len   = axis ? fmaxf(x2 - x1, 1.0f) : fmaxf(y2 - y1, 1.0f);
    const float binsz = len / (float)PP;
    const float g     = ceilf(binsz);   // ceil(len/P) >= 1
    const int   gi    = (int)g;
    const float invg  = 1.0f / g;
    const float wbase = invg * (axis ? scale : 1.0f);

    float* rowp = base + axis*(16*56) + p*56;
    for (int x = 0; x < 56; ++x) rowp[x] = 0.0f;

    for (int s = 0; s < 8; ++s) {
      if (s >= gi) continue;                              // smask: s < g
      const float coord = start + (float)p * binsz + ((float)s + 0.5f) * binsz * invg;
      if (!(coord >= -1.0f && coord <= (float)DIMHW)) continue;  // valid
      const float cc = fmaxf(coord, 0.0f);
      int low = (int)floorf(cc);
      int high; float l;
      if (low >= DIMHW - 1) { low = DIMHW - 1; high = DIMHW - 1; l = 0.0f; }
      else                  { high = low + 1;  l = cc - (float)low; }
      rowp[low]  += (1.0f - l) * wbase;
      rowp[high] += l * wbase;
    }
  } else {
    // zero the pad rows 7..15 of both matrices (2 * 9 * 56 = 1008 floats)
    for (int i = t - 2*PP; i < 2*9*56; i += 64 - 2*PP) {
      const int mat = i / (9*56);
      const int rem = i % (9*56);
      base[mat*(16*56) + (7 + rem/56)*56 + (rem % 56)] = 0.0f;
    }
  }
}

// ---------------------------------------------------------------------------
// Kernel 3: one wave per (b, n, channel-block). For each channel:
//   acc(7x7) = sum over 9 rois of  Ay_r * fm_c * Ax_rT   (scale baked in Ax)
// Stage A: T(16x64) = Ay*fm via 4 N-tiles x 14 K-steps of WMMA f32 16x16x4.
// T -> LDS (stride 68) to convert D-layout to A-layout, then
// Stage B: acc += T * AxT (14 K-steps), accumulated across rois in fixed order.
// ---------------------------------------------------------------------------
__global__ __launch_bounds__(32) void roi_wmma_kernel(
    const float* __restrict__ fmT,
    const float* __restrict__ Amats,
    float* __restrict__ out) {
  const int lane  = threadIdx.x;          // 0..31, wave32
  const int row   = lane & 15;
  const int khalf = (lane >> 4) * 2;      // 0 or 2

  const int wg = blockIdx.x;
  const int cb = wg % CBLK;
  const int bn = wg / CBLK;
  const int b  = bn / NBOX;
  const int n  = bn % NBOX;

  __shared__ __attribute__((aligned(16))) float Tlds[16 * 68];

  for (int cl = 0; cl < CB; ++cl) {
    const int c = cb * CB + cl;
    const float* fmc = fmT + ((size_t)(b*CTOT + c)) * (64*56);

    v8f acc = {};
    for (int rr = 0; rr < 1 + MGT; ++rr) {
      const int r = (rr == 0) ? (b*RPI + n)
                              : (b*RPI + NBOX + n*MGT + (rr - 1));
      const float* Ay = Amats + (size_t)r * AMAT_FL;
      const float* Ax = Ay + 16*56;

      // ---- stage A: T = Ay * fm ----
      v8f t0 = {}, t1 = {}, t2 = {}, t3 = {};
      for (int kk = 0; kk < 14; ++kk) {
        const int kb = 4*kk + khalf;                       // even -> b64 loads
        const v2f a  = *(const v2f*)(Ay + row*56 + kb);    // Ay[M=row][kb..kb+1]
        const v2f b0 = *(const v2f*)(fmc + (row +  0)*56 + kb);  // fmT[w][h..h+1]
        const v2f b1 = *(const v2f*)(fmc + (row + 16)*56 + kb);
        const v2f b2 = *(const v2f*)(fmc + (row + 32)*56 + kb);
        const v2f b3 = *(const v2f*)(fmc + (row + 48)*56 + kb);
        t0 = wmma4(a, b0, t0);
        t1 = wmma4(a, b1, t1);
        t2 = wmma4(a, b2, t2);
        t3 = wmma4(a, b3, t3);
      }

      // ---- spill T to LDS (D-layout -> rows) ----
      const int rbase = (lane < 16) ? 0 : 8;
      #pragma unroll
      for (int jj = 0; jj < 8; ++jj) {
        Tlds[(rbase + jj)*68 + row +  0] = t0[jj];
        Tlds[(rbase + jj)*68 + row + 16] = t1[jj];
        Tlds[(rbase + jj)*68 + row + 32] = t2[jj];
        Tlds[(rbase + jj)*68 + row + 48] = t3[jj];
      }
      __syncthreads();

      // ---- stage B: acc += T * AxT  (B[k][q] = Ax[q][k], scale baked in) ----
      for (int kk = 0; kk < 14; ++kk) {
        const int kb = 4*kk + khalf;
        const v2f a2 = *(const v2f*)(&Tlds[row*68 + kb]);  // T[M=row][kb..kb+1]
        const v2f b2 = *(const v2f*)(Ax + row*56 + kb);    // Ax[q=row][kb..kb+1]
        acc = wmma4(a2, b2, acc);
      }
      __syncthreads();   // WAR: next roi rewrites Tlds
    }

    // acc layout: lane<16 -> vgpr j = D[p=j][q=lane]; p,q valid in 0..6
    if (lane < PP) {
      float* o = out + (((size_t)(b*NBOX + n)*CTOT + c) * (PP*PP));
      #pragma unroll
      for (int jj = 0; jj < PP; ++jj) o[jj*PP + lane] = acc[jj];
    }
  }
}

// ---------------------------------------------------------------------------
extern "C" void kernel_launch(void* const* d_in, const int* in_sizes, int n_in,
                              void* d_out, int out_size, void* d_ws, size_t ws_size,
                              hipStream_t stream) {
  const float* fm    = (const float*)d_in[0];   // [2][256][56][56]
  const float* boxes = (const float*)d_in[1];   // [2][64][4]
  const float* gts   = (const float*)d_in[2];   // [2][8][4]
  float* out = (float*)d_out;                   // [2][64][256][7][7]

  float* fmT   = (float*)d_ws;                  // FMT_FL floats (~7.3 MB)
  float* Amats = fmT + FMT_FL;                  // 1152 * AMAT_FL floats (~8.3 MB)

  transpose_fm_kernel<<<BATCH*CTOT, 256, 0, stream>>>(fm, fmT);
  prep_axes_kernel<<<BATCH*RPI, 64, 0, stream>>>(boxes, gts, Amats);
  roi_wmma_kernel<<<BATCH*NBOX*CBLK, 32, 0, stream>>>(fmT, Amats, out);
}